// KalmanNet_58755152609550
// MI455X (gfx1250) — compile-verified
//
#include <hip/hip_runtime.h>
#include <math.h>
#include <stdint.h>

#define B_    128
#define T_    4096
#define M_    64
#define D_    8
#define NS_   128
#define NW_   30
#define HID_  1024
#define H3_   3072
#define TS_   0.1f
#define BETA_ 0.95f
#define PI_   3.14159265358979f

typedef float v2f __attribute__((ext_vector_type(2)));
typedef float v8f __attribute__((ext_vector_type(8)));

__device__ __forceinline__ float2 c_add(float2 a, float2 b){ return make_float2(a.x+b.x, a.y+b.y); }
__device__ __forceinline__ float2 c_sub(float2 a, float2 b){ return make_float2(a.x-b.x, a.y-b.y); }
__device__ __forceinline__ float2 c_mul(float2 a, float2 b){ return make_float2(a.x*b.x-a.y*b.y, a.x*b.y+a.y*b.x); }
// a * conj(b)
__device__ __forceinline__ float2 c_mulc(float2 a, float2 b){ return make_float2(a.x*b.x+a.y*b.y, a.y*b.x-a.x*b.y); }
__device__ __forceinline__ float2 c_scale(float s, float2 a){ return make_float2(s*a.x, s*a.y); }
__device__ __forceinline__ float2 c_inv(float2 a){ float d=a.x*a.x+a.y*a.y+1e-30f; return make_float2(a.x/d, -a.y/d); }

// ---------------------------------------------------------------------------
// ESPRIT init: per-batch covariance (64x64 Hermitian), dominant-8 subspace via
// orthogonal (subspace) iteration, Phi = pinv(Es[:-1]) Es[1:] via normal eqs,
// eigvals of 8x8 via unshifted QR iteration. One workgroup (64 threads) per b.
// ---------------------------------------------------------------------------
__global__ __launch_bounds__(64) void esprit_init_kernel(
    const float* __restrict__ Xre, const float* __restrict__ Xim,
    float* __restrict__ xst)
{
  __shared__ float2 R[M_][M_];       // 32 KB
  __shared__ float2 Xc[16][M_];      // 8 KB staging
  __shared__ float2 V[M_][D_];
  __shared__ float2 Vn[M_][D_];
  __shared__ float2 G[D_][D_], Cm[D_][D_], Phi[D_][D_], Q[D_][D_], Rt[D_][D_];
  __shared__ float2 sc;
  __shared__ float  sr;
  __shared__ float  th[2][D_];

  const int b = blockIdx.x;
  const int m = threadIdx.x;

  #pragma unroll 1
  for (int win = 0; win < 2; ++win) {
    const int t0 = win * NS_;
    for (int n = 0; n < M_; ++n) R[m][n] = make_float2(0.f, 0.f);
    __syncthreads();

    #pragma unroll 1
    for (int tc = 0; tc < NS_/16; ++tc) {
      #pragma unroll 1
      for (int t = 0; t < 16; ++t) {
        size_t ix = ((size_t)b*T_ + (size_t)(t0 + tc*16 + t))*M_ + m;
        Xc[t][m] = make_float2(Xre[ix], Xim[ix]);
      }
      __syncthreads();
      #pragma unroll 1
      for (int n = 0; n < M_; ++n) {
        float2 acc = make_float2(0.f, 0.f);
        for (int t = 0; t < 16; ++t) acc = c_add(acc, c_mulc(Xc[t][m], Xc[t][n]));
        R[m][n] = c_add(R[m][n], acc);
      }
      __syncthreads();
    }
    for (int n = 0; n < M_; ++n) R[m][n] = c_scale(1.0f/NS_, R[m][n]);

    for (int d = 0; d < D_; ++d)
      V[m][d] = make_float2(sinf(0.7f*m + 1.3f*d) + ((m == d) ? 1.0f : 0.0f),
                            0.1f*cosf(0.3f*m + 0.7f*d));
    __syncthreads();

    #pragma unroll 1
    for (int it = 0; it < 16; ++it) {
      for (int d = 0; d < D_; ++d) {
        float2 acc = make_float2(0.f, 0.f);
        #pragma unroll 1
        for (int n = 0; n < M_; ++n) acc = c_add(acc, c_mul(R[m][n], V[n][d]));
        Vn[m][d] = acc;
      }
      __syncthreads();
      for (int d = 0; d < D_; ++d) V[m][d] = Vn[m][d];
      __syncthreads();
      #pragma unroll 1
      for (int j = 0; j < D_; ++j) {
        if (m == 0) {
          float nr = 0.f;
          for (int r = 0; r < M_; ++r) nr += V[r][j].x*V[r][j].x + V[r][j].y*V[r][j].y;
          sr = rsqrtf(nr + 1e-30f);
        }
        __syncthreads();
        V[m][j] = c_scale(sr, V[m][j]);
        __syncthreads();
        #pragma unroll 1
        for (int k = j + 1; k < D_; ++k) {
          if (m == 0) {
            float2 dd = make_float2(0.f, 0.f);
            for (int r = 0; r < M_; ++r) dd = c_add(dd, c_mulc(V[r][k], V[r][j]));
            sc = dd;
          }
          __syncthreads();
          V[m][k] = c_sub(V[m][k], c_mul(sc, V[m][j]));
          __syncthreads();
        }
      }
    }

    {
      int d = m >> 3, e = m & 7;
      float2 g  = make_float2(0.f, 0.f);
      float2 c2 = make_float2(0.f, 0.f);
      #pragma unroll 1
      for (int r = 0; r < M_ - 1; ++r) {
        g  = c_add(g,  c_mulc(V[r][e],   V[r][d]));
        c2 = c_add(c2, c_mulc(V[r+1][e], V[r][d]));
      }
      G[d][e] = g; Cm[d][e] = c2;
    }
    __syncthreads();

    if (m == 0) {
      #pragma unroll 1
      for (int k = 0; k < D_; ++k) {
        float2 iv = c_inv(G[k][k]);
        for (int j = 0; j < D_; ++j) { G[k][j] = c_mul(G[k][j], iv); Cm[k][j] = c_mul(Cm[k][j], iv); }
        for (int r = 0; r < D_; ++r) {
          if (r == k) continue;
          float2 f = G[r][k];
          for (int j = 0; j < D_; ++j) {
            G[r][j]  = c_sub(G[r][j],  c_mul(f, G[k][j]));
            Cm[r][j] = c_sub(Cm[r][j], c_mul(f, Cm[k][j]));
          }
        }
      }
      for (int i = 0; i < D_; ++i)
        for (int j = 0; j < D_; ++j) Phi[i][j] = Cm[i][j];

      #pragma unroll 1
      for (int it = 0; it < 60; ++it) {
        for (int j = 0; j < D_; ++j) {
          float2 v[D_];
          for (int i = 0; i < D_; ++i) v[i] = Phi[i][j];
          for (int i2 = 0; i2 < j; ++i2) {
            float2 rij = make_float2(0.f, 0.f);
            for (int r = 0; r < D_; ++r) rij = c_add(rij, c_mulc(v[r], Q[r][i2]));
            Rt[i2][j] = rij;
            for (int r = 0; r < D_; ++r) v[r] = c_sub(v[r], c_mul(rij, Q[r][i2]));
          }
          float nr = 0.f;
          for (int r = 0; r < D_; ++r) nr += v[r].x*v[r].x + v[r].y*v[r].y;
          float inr = rsqrtf(nr + 1e-30f);
          Rt[j][j] = make_float2(sqrtf(nr + 1e-30f), 0.f);
          for (int r = 0; r < D_; ++r) Q[r][j] = c_scale(inr, v[r]);
        }
        for (int i = 0; i < D_; ++i)
          for (int j = 0; j < D_; ++j) {
            float2 acc = make_float2(0.f, 0.f);
            for (int r = i; r < D_; ++r) acc = c_add(acc, c_mul(Rt[i][r], Q[r][j]));
            Phi[i][j] = acc;
          }
      }

      float tt[D_];
      for (int d = 0; d < D_; ++d) {
        float2 lam = Phi[d][d];
        float ang = atan2f(lam.y, lam.x);
        float s = -ang / PI_;
        s = fminf(1.f, fmaxf(-1.f, s));
        tt[d] = asinf(s);
      }
      for (int i = 1; i < D_; ++i) {
        float key = tt[i]; int j = i - 1;
        while (j >= 0 && tt[j] > key) { tt[j+1] = tt[j]; --j; }
        tt[j+1] = key;
      }
      for (int d = 0; d < D_; ++d) th[win][d] = tt[d];
    }
    __syncthreads();
  }

  if (m < D_) {
    float t1 = th[1][m], t0v = th[0][m];
    xst[b*24 + m*3 + 0] = t1;
    xst[b*24 + m*3 + 1] = (t1 - t0v) / TS_;
    xst[b*24 + m*3 + 2] = 0.f;
  }
}

// ---------------------------------------------------------------------------
__global__ __launch_bounds__(64) void opast_init_kernel(float2* __restrict__ Wg,
                                                        float2* __restrict__ Pg)
{
  int b = blockIdx.x, m = threadIdx.x;
  for (int d = 0; d < D_; ++d)
    Wg[((size_t)b*M_ + m)*D_ + d] = make_float2((m == d) ? 1.f : 0.f, 0.f);
  int d = m >> 3, e = m & 7;
  Pg[b*64 + m] = make_float2((d == e) ? 0.01f : 0.f, 0.f);
}

// ---------------------------------------------------------------------------
__global__ __launch_bounds__(64) void predict_steer_kernel(
    const float* __restrict__ xst, float* __restrict__ xpred, float* __restrict__ h0)
{
  int b = blockIdx.x, m = threadIdx.x;
  __shared__ float thp[D_];
  if (m < D_) {
    float x0 = xst[b*24 + m*3 + 0];
    float x1 = xst[b*24 + m*3 + 1];
    float x2 = xst[b*24 + m*3 + 2];
    float p0 = x0 + TS_*x1 + 0.5f*TS_*TS_*x2;
    float p1 = x1 + TS_*x2;
    xpred[b*24 + m*3 + 0] = p0;
    xpred[b*24 + m*3 + 1] = p1;
    xpred[b*24 + m*3 + 2] = x2;
    thp[m] = p0;
  }
  __syncthreads();
  for (int d = 0; d < D_; ++d) {
    float ph = PI_ * (float)m * sinf(thp[d]);
    h0[(size_t)b*HID_ + m*D_ + d]       =  cosf(ph);
    h0[(size_t)b*HID_ + 512 + m*D_ + d] = -sinf(ph);
  }
}

// ---------------------------------------------------------------------------
// OPAST recursion: one workgroup per batch, W (64x8) & P (8x8) LDS-resident.
// Per-step sample vector x_t is fetched straight into LDS with the CDNA5
// async global->LDS path (GLOBAL_LOAD_ASYNC_TO_LDS_B32, tracked by ASYNCcnt).
// ---------------------------------------------------------------------------
__global__ __launch_bounds__(64) void opast_kernel(
    const float* __restrict__ Xre, const float* __restrict__ Xim,
    float2* __restrict__ Wg, float2* __restrict__ Pg,
    float* __restrict__ feat, int w)
{
  int b = blockIdx.x, m = threadIdx.x;
  __shared__ float2 W[M_][D_];
  __shared__ float2 P[D_][D_];
  __shared__ float2 xv[M_], yv[D_], qv[D_], pv[M_];
  __shared__ float2 gam;
  __shared__ float  tau_s, c1_s;

  for (int d = 0; d < D_; ++d) W[m][d] = Wg[((size_t)b*M_ + m)*D_ + d];
  P[m >> 3][m & 7] = Pg[b*64 + m];

  const uint32_t lds_re = (uint32_t)(uintptr_t)(&xv[m].x);
  const uint32_t lds_im = (uint32_t)(uintptr_t)(&xv[m].y);
  __syncthreads();

  const int tbase = (2 + w) * NS_;
  #pragma unroll 1
  for (int t = 0; t < NS_; ++t) {
    {
      size_t ix = ((size_t)b*T_ + (size_t)(tbase + t))*M_ + m;
      const float* gre = Xre + ix;
      const float* gim = Xim + ix;
      asm volatile("global_load_async_to_lds_b32 %0, %1, off"
                   :: "v"(lds_re), "v"(gre) : "memory");
      asm volatile("global_load_async_to_lds_b32 %0, %1, off"
                   :: "v"(lds_im), "v"(gim) : "memory");
      asm volatile("s_wait_asynccnt 0" ::: "memory");
    }
    __syncthreads();

    if (m < D_) {                                  // y = W^H x
      float2 acc = make_float2(0.f, 0.f);
      for (int r = 0; r < M_; ++r) acc = c_add(acc, c_mulc(xv[r], W[r][m]));
      yv[m] = acc;
    }
    __syncthreads();

    if (m < D_) {                                  // q = P y / beta
      float2 acc = make_float2(0.f, 0.f);
      for (int e = 0; e < D_; ++e) acc = c_add(acc, c_mul(P[m][e], yv[e]));
      qv[m] = c_scale(1.0f/BETA_, acc);
    }
    __syncthreads();

    if (m == 0) {                                  // gamma = 1 / (1 + y^H q)
      float2 s = make_float2(0.f, 0.f);
      for (int d = 0; d < D_; ++d) s = c_add(s, c_mulc(qv[d], yv[d]));
      gam = c_inv(make_float2(1.f + s.x, s.y));
    }
    __syncthreads();

    {                                              // p = gamma (x - W y)
      float2 acc = make_float2(0.f, 0.f);
      for (int d = 0; d < D_; ++d) acc = c_add(acc, c_mul(W[m][d], yv[d]));
      pv[m] = c_mul(gam, c_sub(xv[m], acc));
    }
    __syncthreads();

    if (m == 0) {                                  // tau, 1 + tau*|q|^2
      float qn2 = 0.f, pn2 = 0.f;
      for (int d = 0; d < D_; ++d) qn2 += qv[d].x*qv[d].x + qv[d].y*qv[d].y;
      for (int r = 0; r < M_; ++r) pn2 += pv[r].x*pv[r].x + pv[r].y*pv[r].y;
      float tau = (1.0f/(qn2 + 1e-20f)) * (rsqrtf(1.0f + pn2*qn2) - 1.0f);
      tau_s = tau; c1_s = 1.0f + tau*qn2;
    }
    __syncthreads();

    {                                              // P = P/beta - gamma q q^H
      int d = m >> 3, e = m & 7;
      float2 qq = c_mul(gam, c_mulc(qv[d], qv[e]));
      P[d][e] = c_sub(c_scale(1.0f/BETA_, P[d][e]), qq);
    }
    __syncthreads();

    {                                              // W += (tau Wq + c1 p) q^H
      float2 wq = make_float2(0.f, 0.f);
      for (int d = 0; d < D_; ++d) wq = c_add(wq, c_mul(W[m][d], qv[d]));
      float2 pp = c_add(c_scale(tau_s, wq), c_scale(c1_s, pv[m]));
      for (int d = 0; d < D_; ++d) W[m][d] = c_add(W[m][d], c_mulc(pp, qv[d]));
    }
    __syncthreads();
  }

  for (int d = 0; d < D_; ++d) {
    Wg[((size_t)b*M_ + m)*D_ + d] = W[m][d];
    feat[(size_t)b*HID_ + m*D_ + d]       = W[m][d].x;
    feat[(size_t)b*HID_ + 512 + m*D_ + d] = W[m][d].y;
  }
  Pg[b*64 + m] = P[m >> 3][m & 7];
}

// ---------------------------------------------------------------------------
// Zero-pad a weight matrix (rows_v x cols_v) into (rows_p x cols_p) so the
// WMMA GEMMs need no bounds guards (no EXEC games around v_wmma).
// ---------------------------------------------------------------------------
__global__ void pad_weights_kernel(const float* __restrict__ src, float* __restrict__ dst,
                                   int rows_v, int cols_v, int rows_p, int cols_p)
{
  int i = blockIdx.x * blockDim.x + threadIdx.x;
  if (i >= rows_p * cols_p) return;
  int r = i / cols_p, c = i % cols_p;
  dst[i] = (r < rows_v && c < cols_v) ? src[r * cols_v + c] : 0.f;
}

// ---------------------------------------------------------------------------
// fp32 WMMA GEMM: Out[128, N] = X[128,K] @ Wt[N,K]^T via V_WMMA_F32_16X16X4_F32.
// Requirements (guaranteed by padding): K % 16 == 0, N % 16 == 0.
// One 16x16 tile per wave32; K unrolled x4 (4 wmma / iter); b64 fragment loads.
// A layout: lanes 0-15 rows with {K0,K1}; lanes 16-31 same rows with {K2,K3}.
// ---------------------------------------------------------------------------
__global__ __launch_bounds__(256) void gemm_wmma_kernel(
    const float* __restrict__ X, int ldx,
    const float* __restrict__ Wt, int ldwt,
    float* __restrict__ Out, int ldo,
    int Ntiles, int K)
{
  const int wave = threadIdx.x >> 5;
  const int lane = threadIdx.x & 31;
  const int tile = blockIdx.x * 8 + wave;
  const int total = 8 * Ntiles;                    // M = 128 -> 8 row tiles
  if (tile >= total) return;                       // wave-uniform exit

  const int tm = tile / Ntiles, tn = tile % Ntiles;
  const int lr = lane & 15, hi = lane >> 4;
  const float2* __restrict__ Xp = (const float2*)(X  + (size_t)(tm*16 + lr)*ldx);
  const float2* __restrict__ Bp = (const float2*)(Wt + (size_t)(tn*16 + lr)*ldwt);

  v8f acc = {0.f, 0.f, 0.f, 0.f, 0.f, 0.f, 0.f, 0.f};
  const int K2 = K >> 1;                           // float2 elems per row

  #pragma unroll 1
  for (int c = hi; c < K2; c += 8) {               // 16 K-values / iteration
    __builtin_prefetch(Bp + c + 16, 0, 3);
    float2 a0 = Xp[c + 0], a1 = Xp[c + 2], a2 = Xp[c + 4], a3 = Xp[c + 6];
    float2 b0 = Bp[c + 0], b1 = Bp[c + 2], b2 = Bp[c + 4], b3 = Bp[c + 6];
    v2f av, bv;
    av.x = a0.x; av.y = a0.y; bv.x = b0.x; bv.y = b0.y;
    acc = __builtin_amdgcn_wmma_f32_16x16x4_f32(false, av, false, bv, (short)0, acc, false, false);
    av.x = a1.x; av.y = a1.y; bv.x = b1.x; bv.y = b1.y;
    acc = __builtin_amdgcn_wmma_f32_16x16x4_f32(false, av, false, bv, (short)0, acc, false, false);
    av.x = a2.x; av.y = a2.y; bv.x = b2.x; bv.y = b2.y;
    acc = __builtin_amdgcn_wmma_f32_16x16x4_f32(false, av, false, bv, (short)0, acc, false, false);
    av.x = a3.x; av.y = a3.y; bv.x = b3.x; bv.y = b3.y;
    acc = __builtin_amdgcn_wmma_f32_16x16x4_f32(false, av, false, bv, (short)0, acc, false, false);
  }

  const int rbase = tm*16 + hi*8;
  const int col   = tn*16 + lr;
  #pragma unroll
  for (int v = 0; v < 8; ++v)
    Out[(size_t)(rbase + v)*ldo + col] = acc[v];
}

// ---------------------------------------------------------------------------
__global__ void bias_u_kernel(float* __restrict__ u, const float* __restrict__ b1)
{
  int i = blockIdx.x * blockDim.x + threadIdx.x;
  if (i >= B_ * 64) return;
  int c = i & 63;
  if (c < 50) u[i] += b1[c];
}

__global__ void gru_gate_kernel(
    const float* __restrict__ gi, const float* __restrict__ gh,
    const float* __restrict__ bih, const float* __restrict__ bhh,
    const float* __restrict__ h0, float* __restrict__ h)
{
  int i = blockIdx.x * blockDim.x + threadIdx.x;
  if (i >= B_ * HID_) return;
  int b = i >> 10, j = i & 1023;
  const float* gib = gi + (size_t)b * H3_;
  const float* ghb = gh + (size_t)b * H3_;
  float gr = gib[j]        + bih[j]        + ghb[j]        + bhh[j];
  float gz = gib[HID_ + j] + bih[HID_ + j] + ghb[HID_ + j] + bhh[HID_ + j];
  float r  = 1.f / (1.f + __expf(-gr));
  float z  = 1.f / (1.f + __expf(-gz));
  float gin = gib[2*HID_ + j] + bih[2*HID_ + j];
  float ghn = ghb[2*HID_ + j] + bhh[2*HID_ + j];
  float n  = tanhf(gin + r * ghn);
  h[i] = (1.f - z) * n + z * h0[i];
}

__global__ __launch_bounds__(32) void fc2_update_kernel(
    const float* __restrict__ delta, const float* __restrict__ b2,
    const float* __restrict__ xpred, float* __restrict__ xst,
    float* __restrict__ out, int w)
{
  int b = blockIdx.x, i = threadIdx.x;
  if (i < 24) {
    float v = xpred[b*24 + i] + delta[b*32 + i] + b2[i];
    xst[b*24 + i] = v;
    if ((i % 3) == 0)
      out[(size_t)b * (NW_ * D_) + (size_t)w * D_ + (i / 3)] = v;
  }
}

// ---------------------------------------------------------------------------
extern "C" void kernel_launch(void* const* d_in, const int* in_sizes, int n_in,
                              void* d_out, int out_size, void* d_ws, size_t ws_size,
                              hipStream_t stream)
{
  (void)in_sizes; (void)n_in; (void)out_size; (void)ws_size;
  const float* Xre   = (const float*)d_in[0];
  const float* Xim   = (const float*)d_in[1];
  const float* fc1_w = (const float*)d_in[2];
  const float* fc1_b = (const float*)d_in[3];
  const float* wih   = (const float*)d_in[4];
  const float* whh   = (const float*)d_in[5];
  const float* bih   = (const float*)d_in[6];
  const float* bhh   = (const float*)d_in[7];
  const float* fc2_w = (const float*)d_in[8];
  const float* fc2_b = (const float*)d_in[9];
  float* out = (float*)d_out;

  float* ws = (float*)d_ws;
  float* xst   = ws; ws += B_ * 24;
  float* xpred = ws; ws += B_ * 24;
  float2* Wg   = (float2*)ws; ws += B_ * M_ * D_ * 2;
  float2* Pg   = (float2*)ws; ws += B_ * 64 * 2;
  float* h0    = ws; ws += B_ * HID_;
  float* feat  = ws; ws += B_ * HID_;
  float* u     = ws; ws += B_ * 64;
  float* gi    = ws; ws += B_ * H3_;
  float* gh    = ws; ws += B_ * H3_;
  float* h     = ws; ws += B_ * HID_;
  float* delta = ws; ws += B_ * 32;
  float* fc1p  = ws; ws += 64  * HID_;   // fc1_w padded 50->64 rows
  float* wihp  = ws; ws += H3_ * 64;     // wih padded K 50->64
  float* fc2p  = ws; ws += 32  * HID_;   // fc2_w padded 24->32 rows

  // one-time weight padding (deterministic, re-done every call)
  pad_weights_kernel<<<(64*HID_ + 255)/256, 256, 0, stream>>>(fc1_w, fc1p, 50, HID_, 64, HID_);
  pad_weights_kernel<<<(H3_*64 + 255)/256, 256, 0, stream>>>(wih,   wihp, H3_, 50, H3_, 64);
  pad_weights_kernel<<<(32*HID_ + 255)/256, 256, 0, stream>>>(fc2_w, fc2p, 24, HID_, 32, HID_);

  esprit_init_kernel<<<B_, 64, 0, stream>>>(Xre, Xim, xst);
  opast_init_kernel<<<B_, 64, 0, stream>>>(Wg, Pg);

  for (int w = 0; w < NW_; ++w) {
    predict_steer_kernel<<<B_, 64, 0, stream>>>(xst, xpred, h0);
    opast_kernel<<<B_, 64, 0, stream>>>(Xre, Xim, Wg, Pg, feat, w);
    // u = feat @ fc1p^T            (M=128, N=64, K=1024), pad cols are zero
    gemm_wmma_kernel<<<4, 256, 0, stream>>>(feat, HID_, fc1p, HID_, u, 64, 4, HID_);
    bias_u_kernel<<<(B_*64 + 255)/256, 256, 0, stream>>>(u, fc1_b);
    // gi = u @ wihp^T              (M=128, N=3072, K=64), zero-padded K
    gemm_wmma_kernel<<<192, 256, 0, stream>>>(u, 64, wihp, 64, gi, H3_, 192, 64);
    // gh = h0 @ whh^T              (M=128, N=3072, K=1024) -- main GEMM
    gemm_wmma_kernel<<<192, 256, 0, stream>>>(h0, HID_, whh, HID_, gh, H3_, 192, HID_);
    gru_gate_kernel<<<(B_*HID_ + 255)/256, 256, 0, stream>>>(gi, gh, bih, bhh, h0, h);
    // delta = h @ fc2p^T           (M=128, N=32, K=1024)
    gemm_wmma_kernel<<<2, 256, 0, stream>>>(h, HID_, fc2p, HID_, delta, 32, 2, HID_);
    fc2_update_kernel<<<B_, 32, 0, stream>>>(delta, fc2_b, xpred, xst, out, w);
  }
}